// GCN_27307402068686
// MI455X (gfx1250) — compile-verified
//
#include <hip/hip_runtime.h>
#include <hip/hip_bf16.h>

typedef __attribute__((ext_vector_type(2))) float v2f;
typedef __attribute__((ext_vector_type(8))) float v8f;

#define N_NODES 50000
#define N_EDGES 800000
#define DIM     128
#define N_LAYERS 4
#define N_GRAPHS 512
#define N_CLASS 10

#define LDS_STRIDE 132   // 128 + 4 pad: kills 64-bank conflicts, keeps 16B alignment

// ---------------------------------------------------------------------------
// WMMA f32 GEMM: out[M,N] = A[M,K=128] @ W[K,N] (+bias) (ReLU optional)
// Block = (N/16) waves; block covers 16 rows x N cols; wave w -> col tile w.
// A strip (16 x 128 f32) staged in LDS once per block via CDNA5 async
// global->LDS (GLOBAL_LOAD_ASYNC_TO_LDS_B128, ASYNCcnt), read back as
// ds_load_b64 fragments. Fragment layouts per CDNA5 ISA 7.12.2:
//   A 16x4:  m = lane&15, kbase = 2*(lane>>4)
//   B 4x16:  n = lane&15, kbase = 2*(lane>>4)
//   C/D:     n = lane&15, row = r + 8*(lane>>4) for acc element r
// ---------------------------------------------------------------------------
__global__ __launch_bounds__(256)
void gemm_wmma_f32(const float* __restrict__ A, const float* __restrict__ W,
                   const float* __restrict__ bias, float* __restrict__ out,
                   int M, int K, int N, int relu_flag)
{
    __shared__ float lds_A[16 * LDS_STRIDE];   // 16 rows x (128 + pad)

    const int tid     = threadIdx.x;
    const int lane    = tid & 31;
    const int wave    = tid >> 5;
    const int rowbase = blockIdx.x * 16;
    const int colbase = wave * 16;
    const int h       = lane >> 4;             // half-wave select
    const int m       = lane & 15;             // A row within tile
    const int n       = m;                     // B/C/D column within tile
    const int kb      = 2 * h;

    // --- stage A strip (16 x K) into LDS with async global->LDS b128 copies ---
    for (int i = tid * 4; i < 16 * K; i += blockDim.x * 4) {
        int r = i >> 7;                        // K = 128
        int c = i & (DIM - 1);
        unsigned     ldsoff = (unsigned)(uintptr_t)(&lds_A[r * LDS_STRIDE + c]);
        const float* gsrc   = A + (size_t)(rowbase + r) * K + c;
        asm volatile("global_load_async_to_lds_b128 %0, %1, off"
                     :: "v"(ldsoff), "v"(gsrc) : "memory");
    }
    asm volatile("s_wait_asynccnt 0x0" ::: "memory");
    __syncthreads();

    // --- K-loop: A fragments from LDS, B fragments streamed from global ---
    v8f acc = {};
    const float* arow = &lds_A[m * LDS_STRIDE];
    for (int k = 0; k < K; k += 4) {
        v2f a, b;
        a.x = arow[k + kb];
        a.y = arow[k + kb + 1];
        b.x = W[(size_t)(k + kb) * N + colbase + n];
        b.y = W[(size_t)(k + kb + 1) * N + colbase + n];
        acc = __builtin_amdgcn_wmma_f32_16x16x4_f32(
            /*neg_a=*/false, a, /*neg_b=*/false, b,
            /*c_mod=*/(short)0, acc, /*reuse_a=*/false, /*reuse_b=*/false);
    }

#pragma unroll
    for (int r = 0; r < 8; ++r) {
        int rr = rowbase + r + 8 * h;
        float v = acc[r];
        if (bias)      v += bias[colbase + n];
        if (relu_flag) v = fmaxf(v, 0.0f);
        out[(size_t)rr * N + colbase + n] = v;
    }
}

// deg[col] += 1 per edge
__global__ void degree_kernel(const int* __restrict__ ei, float* __restrict__ deg, int E)
{
    int e = blockIdx.x * blockDim.x + threadIdx.x;
    if (e < E) unsafeAtomicAdd(&deg[ei[E + e]], 1.0f);
}

__global__ void norm_kernel(const float* __restrict__ deg, float* __restrict__ dinv,
                            float* __restrict__ selfn, int n)
{
    int i = blockIdx.x * blockDim.x + threadIdx.x;
    if (i < n) {
        float d = deg[i] + 1.0f;      // +1 for self loop
        dinv[i]  = rsqrtf(d);
        selfn[i] = 1.0f / d;
    }
}

// one wave per edge: agg[col] += hW[row] * (dinv[row]*dinv[col]),  4 floats/lane
__global__ void scatter_edges(const int* __restrict__ ei, const float* __restrict__ hW,
                              const float* __restrict__ dinv, float* __restrict__ agg, int E)
{
    int wid  = (int)((blockIdx.x * (size_t)blockDim.x + threadIdx.x) >> 5);
    int lane = threadIdx.x & 31;
    if (wid >= E) return;
    int src = ei[wid];
    int dst = ei[E + wid];
    float nrm = dinv[src] * dinv[dst];
    const float* hs = hW + (size_t)src * DIM;
    float*       ad = agg + (size_t)dst * DIM;
#pragma unroll
    for (int j = 0; j < 4; ++j) {
        int c = lane + 32 * j;
        unsafeAtomicAdd(&ad[c], hs[c] * nrm);
    }
}

// agg = relu(agg + hW*selfnorm + bias), in place
__global__ void finalize_kernel(float* __restrict__ agg, const float* __restrict__ hW,
                                const float* __restrict__ selfn, const float* __restrict__ bias,
                                int n)
{
    int idx = blockIdx.x * blockDim.x + threadIdx.x;
    if (idx < n * DIM) {
        int row = idx >> 7;
        int c   = idx & (DIM - 1);
        float v = agg[idx] + hW[idx] * selfn[row] + bias[c];
        agg[idx] = fmaxf(v, 0.0f);
    }
}

// one wave per node: g[batch[i]] += h[i]
__global__ void pool_kernel(const float* __restrict__ h, const int* __restrict__ batch,
                            float* __restrict__ g, int n)
{
    int wid  = (int)((blockIdx.x * (size_t)blockDim.x + threadIdx.x) >> 5);
    int lane = threadIdx.x & 31;
    if (wid >= n) return;
    int b = batch[wid];
    const float* hr = h + (size_t)wid * DIM;
    float*       gr = g + (size_t)b * DIM;
#pragma unroll
    for (int j = 0; j < 4; ++j) {
        int c = lane + 32 * j;
        unsafeAtomicAdd(&gr[c], hr[c]);
    }
}

// out[512,10] = g2[512,64] @ W3[64,10] + b3
__global__ void head_kernel(const float* __restrict__ g2, const float* __restrict__ W3,
                            const float* __restrict__ b3, float* __restrict__ out)
{
    int idx = blockIdx.x * blockDim.x + threadIdx.x;
    if (idx >= N_GRAPHS * N_CLASS) return;
    int gi = idx / N_CLASS;
    int c  = idx % N_CLASS;
    float s = b3[c];
    const float* gr = g2 + gi * (DIM / 2);
#pragma unroll 8
    for (int k = 0; k < DIM / 2; ++k) s += gr[k] * W3[k * N_CLASS + c];
    out[idx] = s;
}

extern "C" void kernel_launch(void* const* d_in, const int* in_sizes, int n_in,
                              void* d_out, int out_size, void* d_ws, size_t ws_size,
                              hipStream_t stream)
{
    const float* x      = (const float*)d_in[0];
    const int*   ei     = (const int*)  d_in[1];
    const int*   batch  = (const int*)  d_in[2];
    const float* W_conv = (const float*)d_in[3];
    const float* b_conv = (const float*)d_in[4];
    const float* W1     = (const float*)d_in[5];
    const float* b1     = (const float*)d_in[6];
    const float* W2     = (const float*)d_in[7];
    const float* b2     = (const float*)d_in[8];
    const float* W3     = (const float*)d_in[9];
    const float* b3     = (const float*)d_in[10];

    // Workspace layout (floats)
    float* bufA  = (float*)d_ws;                    // [N,D] h / agg (in-place)
    float* bufB  = bufA  + (size_t)N_NODES * DIM;   // [N,D] h @ W
    float* deg   = bufB  + (size_t)N_NODES * DIM;   // [N]
    float* dinv  = deg   + N_NODES;                 // [N]
    float* selfn = dinv  + N_NODES;                 // [N]
    float* g     = selfn + N_NODES;                 // [G,128]
    float* g1    = g     + N_GRAPHS * DIM;          // [G,128]
    float* g2    = g1    + N_GRAPHS * DIM;          // [G,64]

    const int T = 256;

    // 1) degrees + normalization coefficients
    hipMemsetAsync(deg, 0, N_NODES * sizeof(float), stream);
    degree_kernel<<<(N_EDGES + T - 1) / T, T, 0, stream>>>(ei, deg, N_EDGES);
    norm_kernel<<<(N_NODES + T - 1) / T, T, 0, stream>>>(deg, dinv, selfn, N_NODES);

    // 2) GCN layers
    for (int l = 0; l < N_LAYERS; ++l) {
        const float* hin = (l == 0) ? x : bufA;
        gemm_wmma_f32<<<N_NODES / 16, 32 * (DIM / 16), 0, stream>>>(
            hin, W_conv + (size_t)l * DIM * DIM, nullptr, bufB, N_NODES, DIM, DIM, 0);
        hipMemsetAsync(bufA, 0, (size_t)N_NODES * DIM * sizeof(float), stream);
        scatter_edges<<<(N_EDGES + 7) / 8, T, 0, stream>>>(ei, bufB, dinv, bufA, N_EDGES);
        finalize_kernel<<<((N_NODES * DIM) + T - 1) / T, T, 0, stream>>>(
            bufA, bufB, selfn, b_conv + (size_t)l * DIM, N_NODES);
    }

    // 3) global add pool
    hipMemsetAsync(g, 0, (size_t)N_GRAPHS * DIM * sizeof(float), stream);
    pool_kernel<<<(N_NODES + 7) / 8, T, 0, stream>>>(bufA, batch, g, N_NODES);

    // 4) MLP head (WMMA for the two 128-K GEMMs, scalar for 64->10)
    gemm_wmma_f32<<<N_GRAPHS / 16, 32 * (DIM / 16), 0, stream>>>(
        g, W1, b1, g1, N_GRAPHS, DIM, DIM, 1);
    gemm_wmma_f32<<<N_GRAPHS / 16, 32 * ((DIM / 2) / 16), 0, stream>>>(
        g1, W2, b2, g2, N_GRAPHS, DIM, DIM / 2, 1);
    head_kernel<<<(N_GRAPHS * N_CLASS + T - 1) / T, T, 0, stream>>>(
        g2, W3, b3, (float*)d_out);
}